// FFTextClassification_77412490543460
// MI455X (gfx1250) — compile-verified
//
#include <hip/hip_runtime.h>
#include <hip/hip_bf16.h>

#define VOCAB  50257
#define NCLS   20
#define BATCH  4096
#define SEQ    200
#define HID    256

typedef __attribute__((ext_vector_type(2))) float v2f;
typedef __attribute__((ext_vector_type(8))) float v8f;

// ---------------------------------------------------------------------------
// Tiled transpose: dst[c*R + r] = src[r*C + c]   (src is [R,C] row-major)
// ---------------------------------------------------------------------------
__global__ void ft_transpose_kernel(const float* __restrict__ src,
                                    float* __restrict__ dst,
                                    int R, int C) {
    __shared__ float tile[32][33];            // +1 pad -> conflict-free
    const int t  = threadIdx.x;               // 256 threads
    const int tx = t & 31;
    const int ty = t >> 5;                    // 0..7
    const int c0 = blockIdx.x * 32;
    const int r0 = blockIdx.y * 32;

    for (int i = ty; i < 32; i += 8) {
        const int r = r0 + i, c = c0 + tx;
        tile[i][tx] = (r < R && c < C) ? src[(size_t)r * C + c] : 0.0f;
    }
    __syncthreads();
    for (int i = ty; i < 32; i += 8) {
        const int c = c0 + i, r = r0 + tx;
        if (c < C && r < R) dst[(size_t)c * R + r] = tile[tx][i];
    }
}

// ---------------------------------------------------------------------------
// Layer 1: y1[b,h] = relu(b1[h] + sum_{unique id in ids[b,:]} W1T[id,h])
// One block (256 threads) per batch row; gathers are coalesced 1KB rows of
// W1T, which lives entirely in the 192MB L2.
// ---------------------------------------------------------------------------
__global__ void ft_layer1_kernel(const int*   __restrict__ ids,
                                 const float* __restrict__ W1T,
                                 const float* __restrict__ b1,
                                 float*       __restrict__ y1) {
    __shared__ int s_ids[SEQ];
    __shared__ int s_keep[SEQ];
    const int b = blockIdx.x;
    const int t = threadIdx.x;                // 0..255 (h index)

    if (t < SEQ) s_ids[t] = ids[(size_t)b * SEQ + t];
    __syncthreads();

    if (t < SEQ) {
        const int v = s_ids[t];
        int keep = 1;
        for (int k = 0; k < t; ++k) {
            if (s_ids[k] == v) { keep = 0; break; }
        }
        s_keep[t] = keep;
    }
    __syncthreads();

    float acc = b1[t];
    for (int j = 0; j < SEQ; ++j) {
        if (s_keep[j]) {                      // uniform per block: no divergence
            acc += W1T[(size_t)s_ids[j] * HID + t];
        }
    }
    y1[(size_t)b * HID + t] = fmaxf(acc, 0.0f);
}

// ---------------------------------------------------------------------------
// Layer 2: y2 = relu(y1 @ W2^T + b2) via V_WMMA_F32_16X16X4_F32.
// Block = 256 threads = 8 waves, covers a 16-row M tile x all 256 outputs.
// Each wave owns two 16x16 N tiles; K=256 consumed in steps of 4.
// A (16x4 f32) layout: lanes 0-15 hold M=0..15 with K={0,1} in v0/v1,
// lanes 16-31 hold K={2,3}.  B mirrors with N on lanes.  C/D: VGPR r holds
// M = r (lanes 0-15) / r+8 (lanes 16-31), N = lane%16.
// ---------------------------------------------------------------------------
__global__ void ft_layer2_wmma_kernel(const float* __restrict__ y1,
                                      const float* __restrict__ W2T,
                                      const float* __restrict__ b2,
                                      float*       __restrict__ y2) {
    __shared__ float As[16][HID + 1];         // pad to break bank conflicts
    const int tid = threadIdx.x;              // 0..255
    const int m0  = blockIdx.x * 16;

    // Stage the 16x256 A tile (coalesced).
    for (int i = 0; i < 16; ++i) {
        As[i][tid] = y1[(size_t)(m0 + i) * HID + tid];
    }
    __syncthreads();

    const int wave  = tid >> 5;               // 0..7
    const int lane  = tid & 31;
    const int lm    = lane & 15;              // M (for A) / N (for B,C,D)
    const int khalf = (lane >> 4) << 1;       // 0 or 2
    const int mbase = m0 + ((lane >> 4) << 3);// C/D row base for this lane half

    for (int nt = 0; nt < 2; ++nt) {
        const int n0 = (wave * 2 + nt) * 16;

        v8f acc;
        const float bias = b2[n0 + lm];       // C[m,n] = b2[n]
        #pragma unroll
        for (int r = 0; r < 8; ++r) acc[r] = bias;

        #pragma unroll 4
        for (int kk = 0; kk < HID; kk += 4) {
            v2f a, bfr;
            a.x   = As[lm][kk + khalf];
            a.y   = As[lm][kk + khalf + 1];
            bfr.x = W2T[(size_t)(kk + khalf) * HID + n0 + lm];
            bfr.y = W2T[(size_t)(kk + khalf + 1) * HID + n0 + lm];
            acc = __builtin_amdgcn_wmma_f32_16x16x4_f32(
                false, a, false, bfr, (short)0, acc, false, false);
        }

        #pragma unroll
        for (int r = 0; r < 8; ++r) {
            const float v = acc[r] > 0.0f ? acc[r] : 0.0f;
            y2[(size_t)(mbase + r) * HID + n0 + lm] = v;
        }
    }
}

// ---------------------------------------------------------------------------
// Layer 3: out[b,o] = b3[o] + y2[b,:] . W3[o,:]   (o < 20)
// One wave per row; 8 rows per 256-thread block.  float4 loads.
// ---------------------------------------------------------------------------
__global__ void ft_layer3_kernel(const float* __restrict__ y2,
                                 const float* __restrict__ W3,
                                 const float* __restrict__ b3,
                                 float*       __restrict__ out) {
    const int row = blockIdx.x * 8 + (threadIdx.x >> 5);
    const int o   = threadIdx.x & 31;
    if (o >= NCLS) return;

    const float4* yr = (const float4*)(y2 + (size_t)row * HID);
    const float4* wr = (const float4*)(W3 + (size_t)o * HID);
    float acc = b3[o];
    #pragma unroll 4
    for (int k = 0; k < HID / 4; ++k) {
        const float4 a = yr[k];
        const float4 w = wr[k];
        acc = fmaf(a.x, w.x, acc);
        acc = fmaf(a.y, w.y, acc);
        acc = fmaf(a.z, w.z, acc);
        acc = fmaf(a.w, w.w, acc);
    }
    out[(size_t)row * NCLS + o] = acc;
}

// ---------------------------------------------------------------------------
extern "C" void kernel_launch(void* const* d_in, const int* in_sizes, int n_in,
                              void* d_out, int out_size, void* d_ws, size_t ws_size,
                              hipStream_t stream) {
    const int*   ids = (const int*)  d_in[0];
    const float* W1  = (const float*)d_in[1];
    const float* b1  = (const float*)d_in[2];
    const float* W2  = (const float*)d_in[3];
    const float* b2  = (const float*)d_in[4];
    const float* W3  = (const float*)d_in[5];
    const float* b3  = (const float*)d_in[6];
    float* out = (float*)d_out;

    // Workspace layout (~60 MB total)
    float* W1T = (float*)d_ws;                        // [VOCAB, HID]
    float* W2T = W1T + (size_t)VOCAB * HID;           // [HID, HID]
    float* y1  = W2T + (size_t)HID * HID;             // [BATCH, HID]
    float* y2  = y1  + (size_t)BATCH * HID;           // [BATCH, HID]

    // 1) W1 [HID, VOCAB] -> W1T [VOCAB, HID]
    {
        dim3 g((VOCAB + 31) / 32, (HID + 31) / 32);
        ft_transpose_kernel<<<g, 256, 0, stream>>>(W1, W1T, HID, VOCAB);
    }
    // 2) W2 [HID, HID] -> W2T [HID, HID]
    {
        dim3 g((HID + 31) / 32, (HID + 31) / 32);
        ft_transpose_kernel<<<g, 256, 0, stream>>>(W2, W2T, HID, HID);
    }
    // 3) Sparse embedding-sum layer (dedup in-block)
    ft_layer1_kernel<<<BATCH, 256, 0, stream>>>(ids, W1T, b1, y1);
    // 4) WMMA GEMM + bias + relu
    ft_layer2_wmma_kernel<<<BATCH / 16, 256, 0, stream>>>(y1, W2T, b2, y2);
    // 5) Final projection
    ft_layer3_kernel<<<BATCH / 8, 256, 0, stream>>>(y2, W3, b3, out);
}